// Ewald_develop_45286135169543
// MI455X (gfx1250) — compile-verified
//
#include <hip/hip_runtime.h>
#include <math.h>

// ---- problem constants (from reference) ----
#define B_SYS   4
#define N_PER   2000
#define K_HEMI  4630                       // # hemisphere k-points (21^3-1)/2
#define KTILE   16
#define NKT     ((K_HEMI + KTILE - 1) / KTILE)   // 290 k-tiles
#define ATILES  (N_PER / 16)               // 125 atom tiles (exact)
#define TWOPI_F 6.28318530717958647692f
#define KSQ_MAX 9.86960440108935861883f    // (2*pi/DL)^2, DL=2 -> pi^2
#define EPS_F   1e-6f
#define NORM_F  90.0474f
#define SELF_DENOM 15.7496099457f          // sigma*(2*pi)^1.5, sigma=1

typedef __attribute__((ext_vector_type(2))) float v2f;
typedef __attribute__((ext_vector_type(8))) float v8f;

// Main kernel: one block = (batch b, tile of 16 k-vectors). 128 threads = 4 waves;
// waves split the 125 atom tiles. Phase matrix 16x16 computed per atom-tile with
// V_WMMA_F32_16X16X4_F32 (A = 16 atoms x [x,y,z,0], B = [x,y,z,0] x 16 kvecs).
__global__ __launch_bounds__(128)
void ewald_sk_kernel(const float* __restrict__ q,
                     const float* __restrict__ r,
                     const float* __restrict__ cell,
                     float* __restrict__ partials)
{
    __shared__ float s_q[N_PER];
    __shared__ float s_r[N_PER * 3];
    __shared__ float s_red[4][2][16];

    const int b     = blockIdx.y;
    const int ktile = blockIdx.x;
    const int tid   = threadIdx.x;
    const int lane  = tid & 31;
    // Wave id is wave-uniform: pin it to an SGPR so the atom-tile loop becomes a
    // scalar s_cbranch loop (no EXEC masking around the WMMA; EXEC stays all-1s
    // as the ISA requires for V_WMMA_*).
    const int wave  = __builtin_amdgcn_readfirstlane(tid >> 5);
    const bool hi   = (lane >= 16);        // upper lane half -> K=2,3 / atom rows 8..15

    // Stage this system's charges and positions into LDS (~32 KB).
    for (int i = tid; i < N_PER; i += 128)     s_q[i] = q[(size_t)b * N_PER + i];
    for (int i = tid; i < N_PER * 3; i += 128) s_r[i] = r[(size_t)b * N_PER * 3 + i];
    __syncthreads();

    // 3x3 inverse of cell (cheap, every thread): G = 2*pi*inv(cell)^T,
    // kvec_e = 2*pi * sum_d n_d * inv(cell)[e][d].
    const float* c = cell + b * 9;
    const float a00=c[0],a01=c[1],a02=c[2],a10=c[3],a11=c[4],a12=c[5],a20=c[6],a21=c[7],a22=c[8];
    const float det = a00*(a11*a22-a12*a21) - a01*(a10*a22-a12*a20) + a02*(a10*a21-a11*a20);
    const float id  = 1.0f / det;
    const float m00=(a11*a22-a12*a21)*id, m01=(a02*a21-a01*a22)*id, m02=(a01*a12-a02*a11)*id;
    const float m10=(a12*a20-a10*a22)*id, m11=(a00*a22-a02*a20)*id, m12=(a02*a10-a00*a12)*id;
    const float m20=(a10*a21-a11*a20)*id, m21=(a01*a20-a00*a21)*id, m22=(a00*a11-a01*a10)*id;

    // This lane's k-vector (column n = lane%16 of the tile).
    // Hemisphere == lexicographic index i in (4630, 9261).
    const int kk = ktile * KTILE + (lane & 15);
    int i = 4631 + kk;
    if (i > 9260) i = 9260;                 // padded ks masked below
    const float n1 = (float)(i / 441 - 10);
    const float n2 = (float)((i / 21) % 21 - 10);
    const float n3 = (float)(i % 21 - 10);
    const float kx = TWOPI_F * (n1*m00 + n2*m01 + n3*m02);
    const float ky = TWOPI_F * (n1*m10 + n2*m11 + n3*m12);
    const float kz = TWOPI_F * (n1*m20 + n2*m21 + n3*m22);

    // B fragment (4x16 f32, KxN). Mirrors ISA A layout:
    // v0: lanes0-15 -> K=0 (kx), lanes16-31 -> K=2 (kz); v1: K=1 (ky) / K=3 (0).
    v2f bf;
    bf.x = hi ? kz : kx;
    bf.y = hi ? 0.0f : ky;

    // Per-lane address select for the A fragment: one unconditional ds_load per
    // component (select the *address*, not between two loads -> v_cndmask, no
    // divergent branch in the hot loop).
    const int comp0 = hi ? 2 : 0;           // x for low half, z for high half

    float acc_re = 0.0f, acc_im = 0.0f;
    for (int t = wave; t < ATILES; t += 4) {
        const int base = t * 16;
        const int row3 = (base + (lane & 15)) * 3;
        const float c0 = s_r[row3 + comp0];
        const float c1 = s_r[row3 + 1];
        v2f af;
        af.x = c0;
        af.y = hi ? 0.0f : c1;
        v8f cz = {};
        // phase tile: D[m][n] = r_m . k_n   (emits v_wmma_f32_16x16x4_f32)
        v8f ph = __builtin_amdgcn_wmma_f32_16x16x4_f32(
            false, af, false, bf, (short)0, cz, false, false);
        // C/D layout: VGPR v -> atom m = base + v + (hi?8:0), col k = lane%16.
        const int mofs = base + (hi ? 8 : 0);
        #pragma unroll
        for (int v = 0; v < 8; ++v) {
            float sn, cs;
            __sincosf(ph[v], &sn, &cs);
            const float qm = s_q[mofs + v];
            acc_re = fmaf(qm, cs, acc_re);
            acc_im = fmaf(qm, sn, acc_im);
        }
    }

    // Fold the two lane halves (lane l and l^16 share the same k column).
    acc_re += __shfl_xor(acc_re, 16, 32);
    acc_im += __shfl_xor(acc_im, 16, 32);
    if (lane < 16) { s_red[wave][0][lane] = acc_re; s_red[wave][1][lane] = acc_im; }
    __syncthreads();

    if (wave == 0) {
        float contrib = 0.0f;
        if (lane < 16) {
            const float sre = s_red[0][0][lane] + s_red[1][0][lane]
                            + s_red[2][0][lane] + s_red[3][0][lane];
            const float sim = s_red[0][1][lane] + s_red[1][1][lane]
                            + s_red[2][1][lane] + s_red[3][1][lane];
            const float ksq = kx*kx + ky*ky + kz*kz;
            const bool ok = (kk < K_HEMI) && (ksq > 0.0f) && (ksq <= KSQ_MAX);
            const float kfac = expf(-0.5f * ksq) / (ksq + EPS_F);   // sigma^2/2 = 0.5
            contrib = ok ? 2.0f * kfac * (sre*sre + sim*sim) : 0.0f;
        }
        #pragma unroll
        for (int off = 8; off; off >>= 1) contrib += __shfl_down(contrib, off, 32);
        // Exactly one writer per (b, ktile): deterministic, no atomics, no ws init.
        if (lane == 0) partials[b * NKT + ktile] = contrib;
    }
}

// Finish: sum k-tile partials, subtract self energy, divide by volume, scale.
__global__ __launch_bounds__(256)
void ewald_finish_kernel(const float* __restrict__ q,
                         const float* __restrict__ cell,
                         const float* __restrict__ partials,
                         float* __restrict__ out)
{
    __shared__ float s_sum[256];
    __shared__ float s_q2[256];
    const int b = blockIdx.x;
    const int tid = threadIdx.x;

    float ksum = 0.0f;
    for (int t = tid; t < NKT; t += 256) ksum += partials[b * NKT + t];
    float q2 = 0.0f;
    for (int i = tid; i < N_PER; i += 256) {
        const float qq = q[(size_t)b * N_PER + i];
        q2 = fmaf(qq, qq, q2);
    }
    s_sum[tid] = ksum; s_q2[tid] = q2;
    __syncthreads();
    for (int s = 128; s; s >>= 1) {
        if (tid < s) { s_sum[tid] += s_sum[tid + s]; s_q2[tid] += s_q2[tid + s]; }
        __syncthreads();
    }
    if (tid == 0) {
        const float* c = cell + b * 9;
        const float vol = c[0]*(c[4]*c[8]-c[5]*c[7])
                        - c[1]*(c[3]*c[8]-c[5]*c[6])
                        + c[2]*(c[3]*c[7]-c[4]*c[6]);
        const float pot = s_sum[0] / vol - s_q2[0] / SELF_DENOM;
        out[b] = pot * NORM_F;
    }
}

extern "C" void kernel_launch(void* const* d_in, const int* in_sizes, int n_in,
                              void* d_out, int out_size, void* d_ws, size_t ws_size,
                              hipStream_t stream) {
    const float* q    = (const float*)d_in[0];   // (B*N, 1) f32
    const float* r    = (const float*)d_in[1];   // (B*N, 3) f32
    const float* cell = (const float*)d_in[2];   // (B, 3, 3) f32
    // d_in[3] = batch ids: sorted equal-size groups -> reshape semantics, unused.
    float* partials = (float*)d_ws;              // B * NKT floats, written before read
    float* out      = (float*)d_out;             // (B,) f32

    dim3 grid(NKT, B_SYS);
    hipLaunchKernelGGL(ewald_sk_kernel, grid, dim3(128), 0, stream, q, r, cell, partials);
    hipLaunchKernelGGL(ewald_finish_kernel, dim3(B_SYS), dim3(256), 0, stream,
                       q, cell, partials, out);
}